// MarkovFlow_49125835931854
// MI455X (gfx1250) — compile-verified
//
#include <hip/hip_runtime.h>
#include <math.h>

typedef __attribute__((ext_vector_type(2))) float v2f;
typedef __attribute__((ext_vector_type(8))) float v8f;

#define L_DIM 128
#define B_DIM 512
#define D_DIM 512
#define S_DIM 128

// ---------------- workspace layout (floats) ----------------
static constexpr size_t OFF_DENS  = 0;                                      // L*B*S = 8388608
static constexpr size_t OFF_MV    = OFF_DENS  + (size_t)L_DIM*B_DIM*S_DIM;  // S*D   = 65536
static constexpr size_t OFF_AEXPT = OFF_MV    + (size_t)S_DIM*D_DIM;        // S*S   = 16384
static constexpr size_t OFF_MSQ   = OFF_AEXPT + (size_t)S_DIM*S_DIM;        // S     = 128
static constexpr size_t OFF_XSQ   = OFF_MSQ   + (size_t)S_DIM;              // L*B   = 65536
static constexpr size_t OFF_LOGC  = OFF_XSQ   + (size_t)L_DIM*B_DIM;        // 1
static constexpr size_t OFF_PART  = OFF_LOGC  + 1;                          // 16

// ---------------- prep: AexpT[j][i] = exp(tparams[i][j] - lse_row_i) ----------------
__global__ void k_prep_trans(const float* __restrict__ tparams, float* __restrict__ AexpT) {
  __shared__ float red[S_DIM];
  int i = blockIdx.x, t = threadIdx.x;
  float v = tparams[i*S_DIM + t];
  red[t] = v; __syncthreads();
  for (int o = S_DIM/2; o > 0; o >>= 1) { if (t < o) red[t] = fmaxf(red[t], red[t+o]); __syncthreads(); }
  float M = red[0]; __syncthreads();
  red[t] = __expf(v - M); __syncthreads();
  for (int o = S_DIM/2; o > 0; o >>= 1) { if (t < o) red[t] += red[t+o]; __syncthreads(); }
  float lse = M + __logf(red[0]);
  AexpT[t*S_DIM + i] = __expf(v - lse);   // transposed store
}

// ---------------- prep: mv[s][d] = means/var, msq[s] = sum means^2/var ----------------
__global__ void k_prep_mv(const float* __restrict__ means, const float* __restrict__ var,
                          float* __restrict__ mv, float* __restrict__ msq) {
  __shared__ float red[256];
  int s = blockIdx.x, t = threadIdx.x;
  float acc = 0.f;
  for (int d = t; d < D_DIM; d += 256) {
    float m = means[s*D_DIM + d];
    float q = m / var[d];
    mv[s*D_DIM + d] = q;
    acc += m * q;
  }
  red[t] = acc; __syncthreads();
  for (int o = 128; o > 0; o >>= 1) { if (t < o) red[t] += red[t+o]; __syncthreads(); }
  if (t == 0) msq[s] = red[0];
}

// ---------------- prep: logc = -(D/2)log(2pi) - 0.5*sum(log var) ----------------
__global__ void k_prep_logc(const float* __restrict__ var, float* __restrict__ logc) {
  __shared__ float red[256];
  int t = threadIdx.x;
  float acc = 0.f;
  for (int d = t; d < D_DIM; d += 256) acc += __logf(var[d]);
  red[t] = acc; __syncthreads();
  for (int o = 128; o > 0; o >>= 1) { if (t < o) red[t] += red[t+o]; __syncthreads(); }
  if (t == 0) logc[0] = -0.5f * (float)D_DIM * 1.8378770664093453f - 0.5f * red[0];
}

// ---------------- prep: xsq[r] = sum_d x[r,d]^2 / var[d]  (one wave per row) ----------------
__global__ void k_prep_xsq(const float* __restrict__ sents, const float* __restrict__ var,
                           float* __restrict__ xsq) {
  int wave = threadIdx.x >> 5, lane = threadIdx.x & 31;
  size_t row = (size_t)blockIdx.x * 8 + wave;
  const float* x = sents + row * D_DIM;
  float acc = 0.f;
  for (int d = lane; d < D_DIM; d += 32) { float xv = x[d]; acc += xv * xv / var[d]; }
  for (int o = 16; o > 0; o >>= 1) acc += __shfl_xor(acc, o, 32);
  if (lane == 0) xsq[row] = acc;
}

// ---------------- density GEMM via V_WMMA_F32_16X16X4_F32 ----------------
// dens[r][s] = logc - 0.5*xsq[r] - 0.5*msq[s] + sum_d X[r,d]*mv[s,d]
__global__ void __launch_bounds__(256) k_density(const float* __restrict__ sents,
                                                 const float* __restrict__ mv,
                                                 const float* __restrict__ msq,
                                                 const float* __restrict__ xsq,
                                                 const float* __restrict__ logcp,
                                                 float* __restrict__ dens) {
  const float logc = logcp[0];
  int wave = threadIdx.x >> 5, lane = threadIdx.x & 31;
  int r0 = (blockIdx.x * 8 + wave) * 16;            // 16-row tile of X (65536 rows)
  int mlane = lane & 15;
  int kk = (lane < 16) ? 0 : 2;                     // f32 A/B fragment K offset per half-wave
  const float* Arow = sents + (size_t)(r0 + mlane) * D_DIM + kk;
  for (int ct = 0; ct < 8; ++ct) {                  // 8 column tiles cover S=128
    int j0 = ct * 16;
    const float* Brow = mv + (size_t)(j0 + mlane) * D_DIM + kk;
    v8f acc = {0.f,0.f,0.f,0.f,0.f,0.f,0.f,0.f};
    for (int k0 = 0; k0 < D_DIM; k0 += 4) {
      v2f a = *(const v2f*)(Arow + k0);
      v2f b = *(const v2f*)(Brow + k0);
      acc = __builtin_amdgcn_wmma_f32_16x16x4_f32(false, a, false, b, (short)0, acc, false, false);
    }
    int ss = j0 + mlane;
    float hmsq = 0.5f * msq[ss];
#pragma unroll
    for (int g = 0; g < 8; ++g) {
      int mm = r0 + g + ((lane < 16) ? 0 : 8);
      dens[(size_t)mm * S_DIM + ss] = logc - 0.5f * xsq[mm] - hmsq + acc[g];
    }
  }
}

// ---------------- persistent forward recursion: one WG per 32 batch rows ----------------
__global__ void __launch_bounds__(256) k_forward(const float* __restrict__ dens,
                                                 const float* __restrict__ AexpT,
                                                 const float* __restrict__ masks,
                                                 float* __restrict__ partial) {
  __shared__ __align__(16) float aS[32 * 132];      // alpha, padded stride
  __shared__ __align__(16) float P [32 * 132];      // exp(alpha - max)
  __shared__ float Mrow[32];
  __shared__ float rowObj[32];
  const int tid  = threadIdx.x;
  const int wave = tid >> 5, lane = tid & 31;
  const int b0   = blockIdx.x * 32;
  const int mlane = lane & 15;
  const int kk    = (lane < 16) ? 0 : 2;
  const float log_pi = -__logf((float)S_DIM);

  // alpha0 = log_pi + dens[l=0]
  for (int idx = tid; idx < 32 * S_DIM; idx += 256) {
    int b = idx >> 7, j = idx & 127;
    aS[b * 132 + j] = log_pi + dens[(size_t)(b0 + b) * S_DIM + j];
  }
  __syncthreads();

  for (int l = 1; l < L_DIM; ++l) {
    // per-row max (8 lanes per row)
    int row = tid >> 3, sub = tid & 7;
    float m = -INFINITY;
    for (int i = sub; i < S_DIM; i += 8) m = fmaxf(m, aS[row * 132 + i]);
    m = fmaxf(m, __shfl_xor(m, 4, 8));
    m = fmaxf(m, __shfl_xor(m, 2, 8));
    m = fmaxf(m, __shfl_xor(m, 1, 8));
    if (sub == 0) Mrow[row] = m;
    __syncthreads();

    // P = exp(alpha - M)
    for (int idx = tid; idx < 32 * S_DIM; idx += 256) {
      int b = idx >> 7, j = idx & 127;
      P[b * 132 + j] = __expf(aS[b * 132 + j] - Mrow[b]);
    }
    __syncthreads();

    // T = P @ Aexp  via WMMA f32; 16 tiles (2 row-tiles x 8 col-tiles), 2 per wave
    for (int t = wave; t < 16; t += 8) {
      int m0 = (t & 1) * 16;
      int j0 = (t >> 1) * 16;
      v8f acc = {0.f,0.f,0.f,0.f,0.f,0.f,0.f,0.f};
      const float* Prow = &P[(m0 + mlane) * 132 + kk];
      const float* Brow = AexpT + (size_t)(j0 + mlane) * S_DIM + kk;
      for (int k0 = 0; k0 < S_DIM; k0 += 4) {
        v2f a = *(const v2f*)(Prow + k0);
        v2f b = *(const v2f*)(Brow + k0);
        acc = __builtin_amdgcn_wmma_f32_16x16x4_f32(false, a, false, b, (short)0, acc, false, false);
      }
      int jj = j0 + mlane;
#pragma unroll
      for (int g = 0; g < 8; ++g) {
        int mm = m0 + g + ((lane < 16) ? 0 : 8);
        float d  = dens[((size_t)l * B_DIM + b0 + mm) * S_DIM + jj];
        float na = d + Mrow[mm] + __logf(acc[g]);
        float mk = masks[l * B_DIM + b0 + mm];
        float od = aS[mm * 132 + jj];
        aS[mm * 132 + jj] = mk * na + (1.f - mk) * od;
      }
    }
    __syncthreads();
  }

  // objective partial: sum_b logsumexp_s alpha[b][s]
  {
    int row = tid >> 3, sub = tid & 7;
    float m = -INFINITY;
    for (int i = sub; i < S_DIM; i += 8) m = fmaxf(m, aS[row * 132 + i]);
    m = fmaxf(m, __shfl_xor(m, 4, 8));
    m = fmaxf(m, __shfl_xor(m, 2, 8));
    m = fmaxf(m, __shfl_xor(m, 1, 8));
    float sm = 0.f;
    for (int i = sub; i < S_DIM; i += 8) sm += __expf(aS[row * 132 + i] - m);
    sm += __shfl_xor(sm, 4, 8);
    sm += __shfl_xor(sm, 2, 8);
    sm += __shfl_xor(sm, 1, 8);
    if (sub == 0) rowObj[row] = m + __logf(sm);
    __syncthreads();
    if (tid == 0) {
      float s = 0.f;
      for (int r = 0; r < 32; ++r) s += rowObj[r];
      partial[blockIdx.x] = s;
    }
  }
}

// ---------------- deterministic final sum ----------------
__global__ void k_final(const float* __restrict__ partial, float* __restrict__ out) {
  if (threadIdx.x == 0 && blockIdx.x == 0) {
    float s = 0.f;
    for (int i = 0; i < B_DIM / 32; ++i) s += partial[i];
    out[0] = s;       // objective
    out[1] = 0.f;     // jacobian_loss (identity transform)
  }
}

extern "C" void kernel_launch(void* const* d_in, const int* in_sizes, int n_in,
                              void* d_out, int out_size, void* d_ws, size_t ws_size,
                              hipStream_t stream) {
  const float* sents   = (const float*)d_in[0];
  const float* masks   = (const float*)d_in[1];
  const float* tparams = (const float*)d_in[2];
  const float* means   = (const float*)d_in[3];
  const float* var     = (const float*)d_in[4];
  float* ws    = (float*)d_ws;
  float* dens  = ws + OFF_DENS;
  float* mv    = ws + OFF_MV;
  float* AexpT = ws + OFF_AEXPT;
  float* msq   = ws + OFF_MSQ;
  float* xsq   = ws + OFF_XSQ;
  float* logc  = ws + OFF_LOGC;
  float* part  = ws + OFF_PART;

  k_prep_trans<<<S_DIM, S_DIM, 0, stream>>>(tparams, AexpT);
  k_prep_mv   <<<S_DIM, 256, 0, stream>>>(means, var, mv, msq);
  k_prep_logc <<<1, 256, 0, stream>>>(var, logc);
  k_prep_xsq  <<<(L_DIM * B_DIM) / 8, 256, 0, stream>>>(sents, var, xsq);
  k_density   <<<(L_DIM * B_DIM) / (16 * 8), 256, 0, stream>>>(sents, mv, msq, xsq, logc, dens);
  k_forward   <<<B_DIM / 32, 256, 0, stream>>>(dens, AexpT, masks, part);
  k_final     <<<1, 32, 0, stream>>>(part, (float*)d_out);
}